// SupplyChainGNN_42769284334133
// MI455X (gfx1250) — compile-verified
//
#include <hip/hip_runtime.h>
#include <math.h>

// Problem constants (match reference)
#define KN      100000
#define KE      1600000
#define KDIN    256
#define KDH     64
#define KDOUT   3
#define KG      64
#define KEPS    1e-5f

typedef __attribute__((ext_vector_type(2))) float v2f;
typedef __attribute__((ext_vector_type(8))) float v8f;

// ---------------------------------------------------------------- degree/norm
__global__ void k_deg_init(float* dinv, int n) {
    int i = blockIdx.x * blockDim.x + threadIdx.x;
    if (i < n) dinv[i] = 1.0f;               // self-loop contributes 1
}

__global__ void k_deg_edges(const int* __restrict__ dst, float* dinv, int e) {
    int i = blockIdx.x * blockDim.x + threadIdx.x;
    if (i < e) atomicAdd(&dinv[dst[i]], 1.0f);
}

__global__ void k_deg_finish(float* dinv, int n) {
    int i = blockIdx.x * blockDim.x + threadIdx.x;
    if (i < n) dinv[i] = rsqrtf(dinv[i]);    // deg >= 1 always
}

__global__ void k_zero(float* p, int n) {
    int i = blockIdx.x * blockDim.x + threadIdx.x;
    if (i < n) p[i] = 0.0f;
}

// ---------------------------------------------------------------- WMMA GEMM
// Y[M x 64] = X[M x K] @ W[K x 64], fp32 via V_WMMA_F32_16X16X4_F32.
// 256 threads = 8 waves; each wave owns one 16x16 output tile
// (2 row-tiles x 4 col-tiles per block). W staged in LDS.
template <int K>
__global__ __launch_bounds__(256) void k_gemm_wmma(const float* __restrict__ X,
                                                   const float* __restrict__ W,
                                                   float* __restrict__ Y,
                                                   int rowTiles) {
    __shared__ float wlds[K * 64];
    for (int t = threadIdx.x; t < K * 64; t += 256) wlds[t] = W[t];
    __syncthreads();

    const int wave = threadIdx.x >> 5;
    const int lane = threadIdx.x & 31;
    const int lm   = lane & 15;                 // M (for A) / N (for B,D) index
    const int koff = (lane >> 4) << 1;          // upper half-wave holds K+2,K+3
    const int rowTile = blockIdx.x * 2 + (wave >> 2);
    const int colBase = (wave & 3) << 4;

    if (rowTile < rowTiles) {                    // wave-uniform guard
        const int rowBase = rowTile << 4;
        const float* __restrict__ arow = X + (size_t)(rowBase + lm) * K;

        v8f acc = {0.f, 0.f, 0.f, 0.f, 0.f, 0.f, 0.f, 0.f};
        for (int k = 0; k < K; k += 4) {
            v2f a, b;
            a.x = arow[k + koff];
            a.y = arow[k + koff + 1];
            b.x = wlds[(k + koff)     * 64 + colBase + lm];
            b.y = wlds[(k + koff + 1) * 64 + colBase + lm];
            acc = __builtin_amdgcn_wmma_f32_16x16x4_f32(
                false, a, false, b, (short)0, acc, false, false);
        }
        // D layout: VGPR v -> M = v (lanes 0-15) / v+8 (lanes 16-31), N = lm
        const int row0 = rowBase + ((lane >> 4) << 3);
#pragma unroll
        for (int v = 0; v < 8; ++v)
            Y[(size_t)(row0 + v) * 64 + colBase + lm] = acc[v];
    }
}

// ------------------------------------------------- propagate (scatter-add)
// out[i,c] = bias[c] + h[i,c]*dinv[i]^2   (self-loop term, full overwrite)
template <int C>
__global__ void k_selfloop_bias(const float* __restrict__ h,
                                const float* __restrict__ dinv,
                                const float* __restrict__ bias,
                                float* __restrict__ out, int n) {
    int idx = blockIdx.x * blockDim.x + threadIdx.x;
    if (idx < n * C) {
        int i = idx / C, c = idx - i * C;
        float d = dinv[i];
        out[idx] = bias[c] + h[idx] * d * d;
    }
}

template <int C>
__global__ void k_edge_prop(const int* __restrict__ src,
                            const int* __restrict__ dst,
                            const float* __restrict__ dinv,
                            const float* __restrict__ h,
                            float* __restrict__ out, int e) {
    long long t = (long long)blockIdx.x * blockDim.x + threadIdx.x;
    if (t < (long long)e * C) {
        int ei = (int)(t / C), c = (int)(t - (long long)ei * C);
        int s = src[ei], d = dst[ei];
        float nrm = dinv[s] * dinv[d];
        atomicAdd(&out[(size_t)d * C + c], h[(size_t)s * C + c] * nrm);
    }
}

// ---------------------------------------------------------------- batch norm
__global__ void k_bn_stats(const float* __restrict__ h,
                           float* __restrict__ stats, int n) {
    __shared__ float ssum[64];
    __shared__ float ssq[64];
    int tid = threadIdx.x;
    if (tid < 64) { ssum[tid] = 0.f; ssq[tid] = 0.f; }
    __syncthreads();
    int c = tid & 63;
    float ls = 0.f, lq = 0.f;
    for (int row = blockIdx.x * 4 + (tid >> 6); row < n; row += gridDim.x * 4) {
        float v = h[(size_t)row * 64 + c];
        ls += v; lq += v * v;
    }
    atomicAdd(&ssum[c], ls);
    atomicAdd(&ssq[c], lq);
    __syncthreads();
    if (tid < 64) {
        atomicAdd(&stats[tid],      ssum[tid]);
        atomicAdd(&stats[64 + tid], ssq[tid]);
    }
}

__global__ void k_bn_relu(float* __restrict__ h,
                          const float* __restrict__ stats,
                          const float* __restrict__ g,
                          const float* __restrict__ be, int n) {
    int idx = blockIdx.x * blockDim.x + threadIdx.x;
    if (idx < n * 64) {
        int c = idx & 63;
        float invn = 1.0f / (float)n;
        float mu  = stats[c] * invn;
        float var = stats[64 + c] * invn - mu * mu;
        float v = g[c] * (h[idx] - mu) * rsqrtf(var + KEPS) + be[c];
        h[idx] = fmaxf(v, 0.0f);
    }
}

// ---------------------------------------------------------------- layer 3
__global__ void k_gemm3(const float* __restrict__ h,
                        const float* __restrict__ W3,
                        float* __restrict__ y, int n) {
    __shared__ float w[KDH * KDOUT];
    if (threadIdx.x < KDH * KDOUT) w[threadIdx.x] = W3[threadIdx.x];
    __syncthreads();
    int i = blockIdx.x * blockDim.x + threadIdx.x;
    if (i < n) {
        const float* hr = h + (size_t)i * KDH;
        float a0 = 0.f, a1 = 0.f, a2 = 0.f;
#pragma unroll 8
        for (int k = 0; k < KDH; ++k) {
            float v = hr[k];
            a0 += v * w[k * 3 + 0];
            a1 += v * w[k * 3 + 1];
            a2 += v * w[k * 3 + 2];
        }
        y[(size_t)i * 3 + 0] = a0;
        y[(size_t)i * 3 + 1] = a1;
        y[(size_t)i * 3 + 2] = a2;
    }
}

// ---------------------------------------------------------------- pool + lsm
__global__ void k_pool(const float* __restrict__ a3,
                       const int* __restrict__ batch,
                       float* __restrict__ pooled,
                       float* __restrict__ cnt, int n) {
    int i = blockIdx.x * blockDim.x + threadIdx.x;
    if (i < n) {
        int gid = batch[i];
        atomicAdd(&pooled[gid * 3 + 0], a3[(size_t)i * 3 + 0]);
        atomicAdd(&pooled[gid * 3 + 1], a3[(size_t)i * 3 + 1]);
        atomicAdd(&pooled[gid * 3 + 2], a3[(size_t)i * 3 + 2]);
        atomicAdd(&cnt[gid], 1.0f);
    }
}

__global__ void k_logsoftmax(const float* __restrict__ pooled,
                             const float* __restrict__ cnt,
                             float* __restrict__ out) {
    int gid = threadIdx.x;
    if (gid < KG) {
        float c = fmaxf(cnt[gid], 1.0f);
        float x0 = pooled[gid * 3 + 0] / c;
        float x1 = pooled[gid * 3 + 1] / c;
        float x2 = pooled[gid * 3 + 2] / c;
        float m = fmaxf(x0, fmaxf(x1, x2));
        float lse = m + logf(expf(x0 - m) + expf(x1 - m) + expf(x2 - m));
        out[gid * 3 + 0] = x0 - lse;
        out[gid * 3 + 1] = x1 - lse;
        out[gid * 3 + 2] = x2 - lse;
    }
}

// ---------------------------------------------------------------- launcher
extern "C" void kernel_launch(void* const* d_in, const int* in_sizes, int n_in,
                              void* d_out, int out_size, void* d_ws, size_t ws_size,
                              hipStream_t stream) {
    const float* x    = (const float*)d_in[0];
    const int*   ei   = (const int*)  d_in[1];
    const int*   bat  = (const int*)  d_in[2];
    const float* W1   = (const float*)d_in[3];
    const float* b1   = (const float*)d_in[4];
    const float* g1   = (const float*)d_in[5];
    const float* be1  = (const float*)d_in[6];
    const float* W2   = (const float*)d_in[7];
    const float* b2   = (const float*)d_in[8];
    const float* g2   = (const float*)d_in[9];
    const float* be2  = (const float*)d_in[10];
    const float* W3   = (const float*)d_in[11];
    const float* b3   = (const float*)d_in[12];

    const int n = in_sizes[0] / KDIN;        // 100000
    const int e = in_sizes[1] / 2;           // 1600000
    const int* src = ei;
    const int* dst = ei + e;

    // workspace layout (floats)
    float* ws     = (float*)d_ws;
    float* dinv   = ws;                                   // n
    float* hA     = dinv   + (size_t)n;                   // n*64 (GEMM out)
    float* hB     = hA     + (size_t)n * 64;              // n*64 (aggregated)
    float* h3     = hB     + (size_t)n * 64;              // n*3
    float* a3     = h3     + (size_t)n * 3;               // n*3
    float* stats1 = a3     + (size_t)n * 3;               // 128
    float* stats2 = stats1 + 128;                         // 128
    float* pooled = stats2 + 128;                         // 192
    float* cnt    = pooled + 192;                         // 64

    const int B = 256;
    const int gN   = (n + B - 1) / B;
    const int gE   = (e + B - 1) / B;
    const int gNC  = (n * 64 + B - 1) / B;
    const long long ec64 = (long long)e * 64;
    const int gEC64 = (int)((ec64 + B - 1) / B);
    const int gEC3  = (int)(((long long)e * 3 + B - 1) / B);
    const int gN3   = (n * 3 + B - 1) / B;
    const int rowTiles = n / 16;                          // 6250
    const int gGemm = (rowTiles + 1) / 2;                 // 2 row-tiles/block

    // norm
    k_deg_init<<<gN, B, 0, stream>>>(dinv, n);
    k_deg_edges<<<gE, B, 0, stream>>>(dst, dinv, e);
    k_deg_finish<<<gN, B, 0, stream>>>(dinv, n);
    k_zero<<<2, B, 0, stream>>>(stats1, 512);   // stats1+stats2+pooled+cnt

    // layer 1
    k_gemm_wmma<KDIN><<<gGemm, B, 0, stream>>>(x, W1, hA, rowTiles);
    k_selfloop_bias<64><<<gNC, B, 0, stream>>>(hA, dinv, b1, hB, n);
    k_edge_prop<64><<<gEC64, B, 0, stream>>>(src, dst, dinv, hA, hB, e);
    k_bn_stats<<<1024, B, 0, stream>>>(hB, stats1, n);
    k_bn_relu<<<gNC, B, 0, stream>>>(hB, stats1, g1, be1, n);

    // layer 2
    k_gemm_wmma<KDH><<<gGemm, B, 0, stream>>>(hB, W2, hA, rowTiles);
    k_selfloop_bias<64><<<gNC, B, 0, stream>>>(hA, dinv, b2, hB, n);
    k_edge_prop<64><<<gEC64, B, 0, stream>>>(src, dst, dinv, hA, hB, e);
    k_bn_stats<<<1024, B, 0, stream>>>(hB, stats2, n);
    k_bn_relu<<<gNC, B, 0, stream>>>(hB, stats2, g2, be2, n);

    // layer 3
    k_gemm3<<<gN, B, 0, stream>>>(hB, W3, h3, n);
    k_selfloop_bias<3><<<gN3, B, 0, stream>>>(h3, dinv, b3, a3, n);
    k_edge_prop<3><<<gEC3, B, 0, stream>>>(src, dst, dinv, h3, a3, e);

    // pool + log_softmax
    k_pool<<<gN, B, 0, stream>>>(a3, bat, pooled, cnt, n);
    k_logsoftmax<<<1, 64, 0, stream>>>(pooled, cnt, (float*)d_out);
}